// DisLoss_65575560675586
// MI455X (gfx1250) — compile-verified
//
#include <hip/hip_runtime.h>
#include <cstdint>
#include <cstddef>

#define FEAT    128
#define GROUP   512              // labels staged per async group (4 x b128 x 32 lanes / 4B)
#define TPG     (GROUP / 32)     // 16 ballot steps per group
#define EMA_A   0.5f
#define EMA_B   0.5f
#define NORMEPS 1e-12f

static __device__ __forceinline__ uint32_t ballot32(int pred) {
#if __has_builtin(__builtin_amdgcn_ballot_w32)
    return __builtin_amdgcn_ballot_w32(pred != 0);
#else
    return (uint32_t)__ballot(pred);
#endif
}

__global__ __launch_bounds__(32)
void disloss_proto_ema(const float* __restrict__ features,
                       const int*   __restrict__ labels,
                       const float* __restrict__ protos_in,
                       float*       __restrict__ protos_out,
                       int nbatch)
{
    const int c    = blockIdx.x;       // one wave32 per class
    const int lane = threadIdx.x;      // 0..31

    __shared__ uint32_t lbuf[2][GROUP];   // double-buffered label stage (4 KB)

    // Prototype row in registers: lane holds elements [4*lane, 4*lane+3]
    float4 p = ((const float4*)(protos_in + (size_t)c * FEAT))[lane];

    // ---- per-match sequential EMA + L2-normalize step ----------------------
    auto process_match = [&](int idx) {
        float4 f = ((const float4*)(features + (size_t)idx * FEAT))[lane];
        float4 v;
        v.x = EMA_A * p.x + EMA_B * f.x;
        v.y = EMA_A * p.y + EMA_B * f.y;
        v.z = EMA_A * p.z + EMA_B * f.z;
        v.w = EMA_A * p.w + EMA_B * f.w;
        float s = v.x * v.x + v.y * v.y + v.z * v.z + v.w * v.w;
        #pragma unroll
        for (int off = 16; off > 0; off >>= 1)
            s += __shfl_xor(s, off, 32);               // wave32 butterfly
        float inv = 1.0f / fmaxf(sqrtf(s), NORMEPS);   // matches max(||v||, eps)
        p.x = v.x * inv; p.y = v.y * inv; p.z = v.z * inv; p.w = v.w * inv;
    };

    // ---- async-stage one group of 512 labels into LDS buffer b -------------
    const uint32_t* glabels = (const uint32_t*)labels;
    auto issue_group = [&](int g, int b) {
        // generic->LDS: low 32 bits of a shared pointer are the group-segment byte offset
        uint32_t ldsbase = (uint32_t)(uintptr_t)(&lbuf[b][0]);
        #pragma unroll
        for (int q = 0; q < 4; ++q) {
            uint32_t voff  = (uint32_t)((g * GROUP + q * 128 + lane * 4) * 4); // global byte off
            uint32_t laddr = ldsbase + (uint32_t)((q * 128 + lane * 4) * 4);   // LDS byte addr
            asm volatile("global_load_async_to_lds_b128 %0, %1, %2"
                         :
                         : "v"(laddr), "v"(voff), "s"(glabels)
                         : "memory");
        }
    };

    // ---- ballot group g's labels; prefetch every matched feature row -------
    auto scan_group = [&](int g, int b, uint32_t (&masks)[TPG]) {
        const int base = g * GROUP;
        #pragma unroll 4
        for (int t = 0; t < TPG; ++t) {
            uint32_t lab = lbuf[b][t * 32 + lane];
            uint32_t m   = ballot32(lab == (uint32_t)c);
            masks[t] = m;
            while (m) {                          // uniform loop
                int bit = __ffs(m) - 1;
                m &= m - 1;
                // lane L touches row byte offset L*16 -> all 4 cachelines of
                // the 512B row prefetched by one global_prefetch_b8
                const float* rp = features
                                + (size_t)(base + t * 32 + bit) * FEAT + lane * 4;
                __builtin_prefetch(rp, 0, 3);
            }
        }
    };

    // ---- walk saved masks in ascending index order -------------------------
    auto process_masks = [&](int g, const uint32_t (&masks)[TPG]) {
        const int base = g * GROUP;
        #pragma unroll 4
        for (int t = 0; t < TPG; ++t) {
            uint32_t mask = masks[t];
            while (mask) {
                int bit = __ffs(mask) - 1;
                mask &= mask - 1;
                process_match(base + t * 32 + bit);
            }
        }
    };

    const int ngroups = nbatch / GROUP;

    if (ngroups > 0) {
        uint32_t masksA[TPG], masksB[TPG];
        issue_group(0, 0);
        for (int g = 0; g < ngroups; ++g) {
            if (g + 1 < ngroups) {
                issue_group(g + 1, (g + 1) & 1);
                // in-order completion: ASYNCcnt<=4 => group g's labels landed
                asm volatile("s_wait_asynccnt 4" ::: "memory");
            } else {
                asm volatile("s_wait_asynccnt 0" ::: "memory");
            }
            // two-deep pipeline: scan+prefetch group g, process group g-1
            if (g & 1) {
                scan_group(g, 1, masksB);
                if (g > 0) process_masks(g - 1, masksA);
            } else {
                scan_group(g, 0, masksA);
                if (g > 0) process_masks(g - 1, masksB);
            }
        }
        if ((ngroups - 1) & 1) process_masks(ngroups - 1, masksB);
        else                   process_masks(ngroups - 1, masksA);
    }

    // ---- scalar tail (nbatch not a multiple of GROUP; unused by harness) ---
    for (int i = ngroups * GROUP; i < nbatch; i += 32) {
        int idx_l = i + lane;
        int lab   = (idx_l < nbatch) ? labels[idx_l] : -1;
        uint32_t mask = ballot32(lab == c);
        while (mask) {
            int bit = __ffs(mask) - 1;
            mask &= mask - 1;
            process_match(i + bit);
        }
    }

    // Every class row is written exactly once (unmatched classes pass through).
    ((float4*)(protos_out + (size_t)c * FEAT))[lane] = p;
}

extern "C" void kernel_launch(void* const* d_in, const int* in_sizes, int n_in,
                              void* d_out, int out_size, void* d_ws, size_t ws_size,
                              hipStream_t stream) {
    const float* features  = (const float*)d_in[0];
    const int*   labels    = (const int*)  d_in[1];
    const float* protos_in = (const float*)d_in[2];
    float*       out       = (float*)d_out;

    const int nbatch = in_sizes[1];        // 131072
    const int ncls   = out_size / FEAT;    // 1000

    disloss_proto_ema<<<dim3(ncls), dim3(32), 0, stream>>>(
        features, labels, protos_in, out, nbatch);
}